// GraphPool_68289980006748
// MI455X (gfx1250) — compile-verified
//
#include <hip/hip_runtime.h>
#include <hip/hip_bf16.h>
#include <math.h>

// ---------------- problem constants ----------------
#define BATCH       4
#define NTOK        4096
#define DDIM        128
#define NUM_CLASSES 5
#define NUM_QUERIES 96
#define SEG         800            // S = (N - 96)/5
#define KK          400            // kk = 0.5 * S
#define NSEL        2096           // 5*400 + 96
#define SORTN       1024           // next pow2 >= SEG

typedef __attribute__((ext_vector_type(2))) float v2f;
typedef __attribute__((ext_vector_type(8))) float v8f;

// ---------------------------------------------------------------------------
// Kernel 1: scores = sigmoid((X @ W + b)/100) via V_WMMA_F32_16X16X4_F32.
// Each wave computes 16 rows. A-tile = X[16 x 4], B-tile = W[k..k+3] broadcast
// into all 16 columns -> every D column holds the same dot product.
// A layout (32b 16x4): lane L (M=L%16): VGPR0 = K=2*(L/16), VGPR1 = K+1.
// B layout (32b 4x16): VGPR0 = row K=2*(L/16), VGPR1 = row K+1 (value
// independent of N since we broadcast W).
// C/D layout: VGPR r, lane L -> M = r + 8*(L/16), N = L%16.
// ---------------------------------------------------------------------------
__global__ __launch_bounds__(256)
void gp_scores_wmma(const float* __restrict__ X, const float* __restrict__ W,
                    const float* __restrict__ bias, float* __restrict__ scores) {
    const int lane = threadIdx.x & 31;
    const int wave = threadIdx.x >> 5;
    const int tile = blockIdx.x * 8 + wave;            // 1024 tiles total
    const int row0 = tile * 16;                        // flat row in (B*N)
    const int half = lane >> 4;                        // 0: lanes 0-15, 1: 16-31
    const int m    = lane & 15;

    const float* __restrict__ xrow = X + (size_t)(row0 + m) * DDIM;
    const float b0 = bias[0];

    v8f c = {};
    #pragma unroll 8
    for (int k0 = 0; k0 < DDIM; k0 += 4) {
        const int ka = k0 + 2 * half;
        v2f a;  a.x = xrow[ka];  a.y = xrow[ka + 1];
        v2f bm; bm.x = W[ka];    bm.y = W[ka + 1];
        // D = A(16x4) * B(4x16) + C   -> v_wmma_f32_16x16x4_f32
        c = __builtin_amdgcn_wmma_f32_16x16x4_f32(
                /*neg_a=*/false, a, /*neg_b=*/false, bm,
                /*c_mod=*/(short)0, c, /*reuse_a=*/false, /*reuse_b=*/false);
    }

    // lanes with N==0 (lane 0 and lane 16) hold rows r+8*half in c[r]
    if ((lane & 15) == 0) {
        #pragma unroll
        for (int r = 0; r < 8; ++r) {
            const int row = row0 + r + 8 * half;
            const float logit = c[r] + b0;
            scores[row] = 1.0f / (1.0f + __expf(-logit * 0.01f));
        }
    }
}

// ---------------------------------------------------------------------------
// Kernel 2: per (batch, class) stable top-KK of SEG scores via LDS bitonic
// sort of SORTN padded (val,idx) pairs. Order: value desc, tie -> index asc
// (matches jax.lax.top_k).
// ---------------------------------------------------------------------------
__global__ __launch_bounds__(512)
void gp_topk(const float* __restrict__ scores, int* __restrict__ idx_all,
             float* __restrict__ vals_all) {
    __shared__ float sv[SORTN];
    __shared__ int   si[SORTN];

    const int b = blockIdx.x / NUM_CLASSES;
    const int c = blockIdx.x % NUM_CLASSES;
    const float* __restrict__ sc = scores + b * NTOK + c * SEG;

    for (int i = threadIdx.x; i < SORTN; i += 512) {
        if (i < SEG) { sv[i] = sc[i]; si[i] = i; }
        else         { sv[i] = -INFINITY; si[i] = 0x7FFFFFFF; }
    }
    __syncthreads();

    for (int k = 2; k <= SORTN; k <<= 1) {
        for (int j = k >> 1; j > 0; j >>= 1) {
            for (int i = threadIdx.x; i < SORTN; i += 512) {
                const int p = i ^ j;
                if (p > i) {
                    const bool up = ((i & k) == 0);     // desired order block
                    const float va = sv[i], vb = sv[p];
                    const int   ia = si[i], ib = si[p];
                    const bool aFirst = (va > vb) || (va == vb && ia < ib);
                    if (up ? !aFirst : aFirst) {
                        sv[i] = vb; sv[p] = va;
                        si[i] = ib; si[p] = ia;
                    }
                }
            }
            __syncthreads();
        }
    }

    for (int t = threadIdx.x; t < KK; t += 512) {
        const int pos = b * NSEL + c * KK + t;
        idx_all[pos]  = si[t] + c * SEG;
        vals_all[pos] = sv[t];
    }
}

// ---------------------------------------------------------------------------
// Kernel 3: append the NUM_QUERIES query slots per batch.
// ---------------------------------------------------------------------------
__global__ void gp_queries(const float* __restrict__ scores,
                           int* __restrict__ idx_all, float* __restrict__ vals_all) {
    const int t = blockIdx.x * blockDim.x + threadIdx.x;
    if (t >= BATCH * NUM_QUERIES) return;
    const int b = t / NUM_QUERIES, q = t % NUM_QUERIES;
    const int pos = b * NSEL + NUM_CLASSES * KK + q;
    idx_all[pos]  = (NTOK - NUM_QUERIES) + q;
    vals_all[pos] = scores[b * NTOK + (NTOK - NUM_QUERIES) + q];
}

// ---------------------------------------------------------------------------
// Kernel 4: new_X[b,i,:] = X[b,idx,:] * val ; also emit idx as float output.
// One block (128 threads) per selected row -> fully coalesced 512B rows.
// ---------------------------------------------------------------------------
__global__ __launch_bounds__(128)
void gp_newx(const float* __restrict__ X, const int* __restrict__ idx_all,
             const float* __restrict__ vals_all, float* __restrict__ out_newX,
             float* __restrict__ out_idx) {
    const int r  = blockIdx.x;                 // 0 .. B*NSEL-1
    const int b  = r / NSEL;
    const int gi = idx_all[r];
    const float v = vals_all[r];
    const float* __restrict__ src = X + ((size_t)b * NTOK + gi) * DDIM;
    float* __restrict__ dst = out_newX + (size_t)r * DDIM;
    dst[threadIdx.x] = src[threadIdx.x] * v;
    if (threadIdx.x == 0) out_idx[r] = (float)gi;
}

// ---------------------------------------------------------------------------
// Kernel 5: new_A[b,i,j] = A[b, idx_i, idx_j]. One block per output row; the
// 2096-entry column-index vector is staged in LDS, row base is uniform per
// block so gathers stay inside one 16KB row of A (L2-friendly).
// ---------------------------------------------------------------------------
__global__ __launch_bounds__(256)
void gp_newa(const float* __restrict__ A, const int* __restrict__ idx_all,
             float* __restrict__ out_newA) {
    __shared__ int sidx[NSEL];
    const int r = blockIdx.x;                  // 0 .. B*NSEL-1
    const int b = r / NSEL;
    const int i = r % NSEL;
    const int* __restrict__ idxrow = idx_all + b * NSEL;
    for (int t = threadIdx.x; t < NSEL; t += 256) sidx[t] = idxrow[t];
    __syncthreads();

    const int gi = sidx[i];
    const float* __restrict__ arow = A + ((size_t)b * NTOK + gi) * NTOK;
    float* __restrict__ dst = out_newA + (size_t)r * NSEL;
    for (int t = threadIdx.x; t < NSEL; t += 256)
        dst[t] = arow[sidx[t]];
}

// ---------------------------------------------------------------------------
extern "C" void kernel_launch(void* const* d_in, const int* in_sizes, int n_in,
                              void* d_out, int out_size, void* d_ws, size_t ws_size,
                              hipStream_t stream) {
    const float* A  = (const float*)d_in[0];   // (4, 4096, 4096)
    const float* X  = (const float*)d_in[1];   // (4, 4096, 128)
    const float* W  = (const float*)d_in[2];   // (128, 1)
    const float* bb = (const float*)d_in[3];   // (1,)

    // output layout: new_A | new_X | idx(as float)
    float* out_newA = (float*)d_out;
    float* out_newX = out_newA + (size_t)BATCH * NSEL * NSEL;
    float* out_idx  = out_newX + (size_t)BATCH * NSEL * DDIM;

    // workspace carve-out
    char* ws = (char*)d_ws;
    float* ws_scores = (float*)(ws);                               // 16384 f
    int*   ws_idx    = (int*)  (ws + 65536);                       //  8384 i
    float* ws_vals   = (float*)(ws + 65536 + 33536);               //  8384 f

    // 1) scores via WMMA: 1024 row-tiles of 16, 8 waves/block
    gp_scores_wmma<<<128, 256, 0, stream>>>(X, W, bb, ws_scores);

    // 2) top-400 per (batch, class): 20 blocks
    gp_topk<<<BATCH * NUM_CLASSES, 512, 0, stream>>>(ws_scores, ws_idx, ws_vals);

    // 3) query slots
    gp_queries<<<(BATCH * NUM_QUERIES + 255) / 256, 256, 0, stream>>>(
        ws_scores, ws_idx, ws_vals);

    // 4) new_X + idx output
    gp_newx<<<BATCH * NSEL, 128, 0, stream>>>(X, ws_idx, ws_vals,
                                              out_newX, out_idx);

    // 5) new_A 2D gather (dominant, memory-bound)
    gp_newa<<<BATCH * NSEL, 256, 0, stream>>>(A, ws_idx, out_newA);
}